// Smooth_Blk_Edges_72962904425121
// MI455X (gfx1250) — compile-verified
//
#include <hip/hip_runtime.h>

// x: (B=2, H=1024, W=1024, C=32) fp32, 16x16 pixel tiles.
// One thread = one float4 (4 channels) of one pixel. 8 threads per pixel.
// Memory-bound: 256 MB in + 256 MB out -> ~22 us floor at 23.3 TB/s.

typedef float v4f __attribute__((ext_vector_type(4)));

#define HH 1024
#define WW 1024

__device__ __forceinline__ v4f XV(const v4f* __restrict__ x, int b, int h, int w, int c4) {
    // idx = (((b*1024 + h)*1024 + w)*8 + c4)   (fits in int: max 2^24)
    return x[(((((b << 10) + h) << 10) + w) << 3) + c4];
}

__global__ __launch_bounds__(256)
void smooth_blk_edges_kernel(const v4f* __restrict__ x, v4f* __restrict__ out, int n) {
    int tid = blockIdx.x * 256 + threadIdx.x;
    if (tid >= n) return;

    // Streaming prefetch ~1 MiB ahead on the input (gfx1250 global_prefetch_b8).
    if (tid + (1 << 16) < n)
        __builtin_prefetch((const void*)(x + tid + (1 << 16)), 0, 1);

    int c4 = tid & 7;
    int w  = (tid >> 3)  & (WW - 1);
    int h  = (tid >> 13) & (HH - 1);
    int b  = tid >> 23;

    int rh = h & 15, rw = w & 15;
    bool hB = (rh == 15) & (h < HH - 1);   // bottom edge row of a tile (pairs with h+1)
    bool hT = (rh == 0)  & (h > 0);        // top edge row (pairs with h-1)
    bool wB = (rw == 15) & (w < WW - 1);
    bool wT = (rw == 0)  & (w > 0);
    bool hE = hB | hT;
    bool wE = wB | wT;

    v4f r;
    if (hE & wE) {
        // Corner quad: rows {hb, hb+1} x cols {wb, wb+1}; 4 raw loads cover everything.
        int hb = hB ? h : h - 1;
        int wb = wB ? w : w - 1;
        v4f xa = XV(x, b, hb,     wb,     c4);
        v4f xb = XV(x, b, hb + 1, wb,     c4);
        v4f xc = XV(x, b, hb,     wb + 1, c4);
        v4f xd = XV(x, b, hb + 1, wb + 1, c4);
        // step 1: vertical pair smoothing (a=1 -> (2u+v)/3, (u+2v)/3)
        v4f va = (2.0f * xa + xb) / 3.0f;
        v4f vb = (xa + 2.0f * xb) / 3.0f;
        v4f vc = (2.0f * xc + xd) / 3.0f;
        v4f vd = (xc + 2.0f * xd) / 3.0f;
        // step 2: horizontal pair smoothing of the vertically smoothed values
        v4f za = (2.0f * va + vc) / 3.0f;   // (hb,   wb)
        v4f zc = (va + 2.0f * vc) / 3.0f;   // (hb,   wb+1)
        v4f zb = (2.0f * vb + vd) / 3.0f;   // (hb+1, wb)
        v4f zd = (vb + 2.0f * vd) / 3.0f;   // (hb+1, wb+1)
        // step 3: 4-point corner correction (a=1 -> d=5, a2+d=6)
        if (hB & wB)      r = ( 6.0f * za -        zb -        zc +        zd) / 5.0f;
        else if (hT & wB) r = (-za        + 6.0f * zb +        zc -        zd) / 5.0f;
        else if (hB & wT) r = (-za        +        zb + 6.0f * zc -        zd) / 5.0f;
        else              r = ( za        -        zb -        zc + 6.0f * zd) / 5.0f;
    } else if (hE | wE) {
        // Exactly one boundary: (2*self + neighbor)/3 from either side.
        v4f x0 = XV(x, b, h, w, c4);
        int hn = hE ? (hB ? h + 1 : h - 1) : h;
        int wn = wE ? (wB ? w + 1 : w - 1) : w;
        v4f xn = XV(x, b, hn, wn, c4);
        r = (2.0f * x0 + xn) / 3.0f;
    } else {
        // Interior (~77% of pixels): pure copy.
        r = XV(x, b, h, w, c4);
    }

    // Output is never re-read: non-temporal store keeps L2 free for input reuse.
    __builtin_nontemporal_store(r, out + tid);
}

extern "C" void kernel_launch(void* const* d_in, const int* in_sizes, int n_in,
                              void* d_out, int out_size, void* d_ws, size_t ws_size,
                              hipStream_t stream) {
    const v4f* x = (const v4f*)d_in[0];
    v4f* out = (v4f*)d_out;
    int n = in_sizes[0] / 4;                 // number of float4 elements
    int block = 256;                         // 8 wave32 waves per block
    int grid = (n + block - 1) / block;
    smooth_blk_edges_kernel<<<grid, block, 0, stream>>>(x, out, n);
}